// SoftSemanticLoss_6846177869930
// MI455X (gfx1250) — compile-verified
//
#include <hip/hip_runtime.h>
#include <hip/hip_bf16.h>
#include <math.h>

typedef __attribute__((ext_vector_type(2))) float v2f;
typedef __attribute__((ext_vector_type(8))) float v8f;

#define Bsz   32
#define Cch   256
#define Wfd   64
#define Mpix  4096          // 64*64
#define Nkp   128
#define NPAIR (Bsz*Nkp)     // 4096
#define INV_TEMP 10.0f
#define EPSN  1e-12f

// ---------------- K1: per-pixel inverse channel L2 norm ----------------
__global__ __launch_bounds__(256) void k_invnorm(const float* __restrict__ feats,
                                                 float* __restrict__ inv) {
    int p = blockIdx.x * blockDim.x + threadIdx.x;     // 0 .. B*M-1
    if (p >= Bsz * Mpix) return;
    int b = p >> 12, m = p & (Mpix - 1);
    const float* base = feats + (size_t)b * Cch * Mpix + m;
    float acc = 0.f;
    #pragma unroll 8
    for (int c = 0; c < Cch; ++c) { float v = base[(size_t)c * Mpix]; acc += v * v; }
    inv[p] = 1.0f / fmaxf(sqrtf(acc), EPSN);
}

// ---------------- K2: per-keypoint prep (bilinear weights / indices) ----------------
__global__ __launch_bounds__(256) void k_prep(const float* __restrict__ kps_src,
                                              const float* __restrict__ kps_trg,
                                              const int*   __restrict__ kps_mask,
                                              const float* __restrict__ inv_src,
                                              float4* __restrict__ srcW, int4* __restrict__ srcIdx,
                                              float4* __restrict__ trgW, int4* __restrict__ trgIdx) {
    int g = blockIdx.x * blockDim.x + threadIdx.x;
    if (g >= NPAIR) return;
    int b = g >> 7;
    // source sampling: x = kx/(1023)*63 ; normalize folded into weights
    float x = kps_src[2*g]     * (63.0f / 1023.0f);
    float y = kps_src[2*g + 1] * (63.0f / 1023.0f);
    int x0 = (int)fminf(fmaxf(floorf(x), 0.f), 63.f);
    int y0 = (int)fminf(fmaxf(floorf(y), 0.f), 63.f);
    int x1 = min(x0 + 1, 63), y1 = min(y0 + 1, 63);
    float fx = x - (float)x0, fy = y - (float)y0;
    int p00 = y0*Wfd + x0, p10 = y1*Wfd + x0, p01 = y0*Wfd + x1, p11 = y1*Wfd + x1;
    const float* invb = inv_src + b * Mpix;
    float4 sw;
    sw.x = (1.f - fx) * (1.f - fy) * invb[p00];
    sw.y = (1.f - fx) * fy         * invb[p10];
    sw.z = fx * (1.f - fy)         * invb[p01];
    sw.w = fx * fy                 * invb[p11];
    srcW[g]   = sw;
    srcIdx[g] = make_int4(p00, p10, p01, p11);
    // target soft-label weights (mask folded in; reference allows negative w at edges)
    float gx = kps_trg[2*g]     * (1.0f / 16.0f);
    float gy = kps_trg[2*g + 1] * (1.0f / 16.0f);
    int tx0 = (int)fminf(fmaxf(floorf(gx), 0.f), 63.f);
    int ty0 = (int)fminf(fmaxf(floorf(gy), 0.f), 63.f);
    int tx1 = min(tx0 + 1, 63), ty1 = min(ty0 + 1, 63);
    float x0f = (float)tx0, x1f = (float)tx1, y0f = (float)ty0, y1f = (float)ty1;
    float mk = (kps_mask[g] != 0) ? 1.f : 0.f;
    float4 tw;
    tw.x = (x1f - gx) * (y1f - gy) * mk;
    tw.y = (x1f - gx) * (gy - y0f) * mk;
    tw.z = (gx - x0f) * (y1f - gy) * mk;
    tw.w = (gx - x0f) * (gy - y0f) * mk;
    trgW[g]   = tw;
    trgIdx[g] = make_int4(ty0*Wfd + tx0, ty1*Wfd + tx0, ty0*Wfd + tx1, ty1*Wfd + tx1);
}

// ---------------- K3: build normalized query features q[B*N][C] ----------------
__global__ __launch_bounds__(256) void k_buildq(const float* __restrict__ feats_src,
                                                const float4* __restrict__ srcW,
                                                const int4*   __restrict__ srcIdx,
                                                float* __restrict__ q) {
    int g = blockIdx.x;          // keypoint pair
    int c = threadIdx.x;         // channel
    int b = g >> 7;
    const float* fb = feats_src + (size_t)b * Cch * Mpix + (size_t)c * Mpix;
    float4 w = srcW[g]; int4 id = srcIdx[g];
    q[(size_t)g * Cch + c] = w.x*fb[id.x] + w.y*fb[id.y] + w.z*fb[id.z] + w.w*fb[id.w];
}

// ---------------- K4: WMMA GEMM (16 rows x 4096 cols, K=256) + fused logsumexp ----------------
__global__ __launch_bounds__(256) void k_gemm_lse(const float* __restrict__ feats_trg,
                                                  const float* __restrict__ inv_trg,
                                                  const float* __restrict__ q,
                                                  float* __restrict__ lse_out) {
    __shared__ float qs[16][258];      // row-block of q, padded against bank conflicts
    __shared__ float pmx[8][16];
    __shared__ float psm[8][16];
    int wg  = blockIdx.x;              // 0..255  (b*8 + row-block)
    int b   = wg >> 3;
    int rb  = wg & 7;
    int tid = threadIdx.x;
    #pragma unroll
    for (int i = 0; i < 16; ++i)       // coalesced stage: 16 rows x 256 channels
        qs[i][tid] = q[(size_t)(b * Nkp + rb * 16 + i) * Cch + tid];
    __syncthreads();

    int wave = tid >> 5, lane = tid & 31;
    int half = lane >> 4, lrow = lane & 15;
    int kb   = 2 * half;               // f32 16x16x4 A/B layout: half 0 -> K {0,1}, half 1 -> K {2,3}
    const float* Tb = feats_trg + (size_t)b * Cch * Mpix;

    float mx[8], sm[8];
    #pragma unroll
    for (int r = 0; r < 8; ++r) { mx[r] = -INFINITY; sm[r] = 0.f; }

    for (int t = wave; t < Mpix / 16; t += 8) {     // 32 column tiles per wave
        int m0 = t * 16;
        v8f acc = {};
        const float* Bp = Tb + m0 + lrow;
        #pragma unroll 4
        for (int k0 = 0; k0 < Cch; k0 += 4) {
            v2f a, bv;
            a.x  = qs[lrow][k0 + kb];
            a.y  = qs[lrow][k0 + kb + 1];
            bv.x = Bp[(size_t)(k0 + kb) * Mpix];
            bv.y = Bp[(size_t)(k0 + kb + 1) * Mpix];
            acc = __builtin_amdgcn_wmma_f32_16x16x4_f32(false, a, false, bv,
                                                        (short)0, acc, false, false);
        }
        // fold target normalization + 1/TEMP into accumulator (column m fixed per lane)
        float scale = inv_trg[b * Mpix + m0 + lrow] * INV_TEMP;
        #pragma unroll
        for (int r = 0; r < 8; ++r) {               // online softmax per (row, lane)
            float x = acc[r] * scale;
            if (x > mx[r]) { sm[r] *= __expf(mx[r] - x); mx[r] = x; }
            sm[r] += __expf(x - mx[r]);
        }
    }
    // combine (mx, s) across the 16 lanes of each half (same row, different columns)
    #pragma unroll
    for (int off = 1; off < 16; off <<= 1) {
        #pragma unroll
        for (int r = 0; r < 8; ++r) {
            float omx = __shfl_xor(mx[r], off, 32);
            float osm = __shfl_xor(sm[r], off, 32);
            float nmx = fmaxf(mx[r], omx);
            sm[r] = sm[r] * __expf(mx[r] - nmx) + osm * __expf(omx - nmx);
            mx[r] = nmx;
        }
    }
    if (lrow == 0) {                   // VGPR r holds row r (half 0) / r+8 (half 1)
        #pragma unroll
        for (int r = 0; r < 8; ++r) { pmx[wave][r + 8*half] = mx[r]; psm[wave][r + 8*half] = sm[r]; }
    }
    __syncthreads();
    if (tid < 16) {                    // combine the 8 wave partials per row
        float mm = -INFINITY, ss = 0.f;
        #pragma unroll
        for (int w = 0; w < 8; ++w) {
            float wm = pmx[w][tid], wsv = psm[w][tid];
            float nm = fmaxf(mm, wm);
            ss = ss * __expf(mm - nm) + wsv * __expf(wm - nm);
            mm = nm;
        }
        lse_out[b * Nkp + rb * 16 + tid] = mm + __logf(ss);
    }
}

// ---------------- K5: 4 sparse target logits + per-keypoint loss ----------------
__global__ __launch_bounds__(256) void k_loss(const float* __restrict__ feats_trg,
                                              const float* __restrict__ inv_trg,
                                              const float* __restrict__ q,
                                              const float* __restrict__ lse,
                                              const float4* __restrict__ trgW,
                                              const int4*   __restrict__ trgIdx,
                                              float* __restrict__ loss) {
    int tid  = threadIdx.x;
    int g    = blockIdx.x * 8 + (tid >> 5);   // one wave per keypoint
    int lane = tid & 31;
    int b    = g >> 7;
    const float* qrow = q + (size_t)g * Cch;
    const float* Tb   = feats_trg + (size_t)b * Cch * Mpix;
    int4 id = trgIdx[g];
    float d0 = 0.f, d1 = 0.f, d2 = 0.f, d3 = 0.f;
    for (int c = lane; c < Cch; c += 32) {
        float qc = qrow[c];
        const float* row = Tb + (size_t)c * Mpix;
        d0 += qc * row[id.x]; d1 += qc * row[id.y];
        d2 += qc * row[id.z]; d3 += qc * row[id.w];
    }
    #pragma unroll
    for (int off = 16; off >= 1; off >>= 1) {
        d0 += __shfl_xor(d0, off, 32);
        d1 += __shfl_xor(d1, off, 32);
        d2 += __shfl_xor(d2, off, 32);
        d3 += __shfl_xor(d3, off, 32);
    }
    if (lane == 0) {
        float4 w = trgW[g];
        const float* inv = inv_trg + b * Mpix;
        float L  = lse[g];
        float l0 = d0 * inv[id.x] * INV_TEMP;
        float l1 = d1 * inv[id.y] * INV_TEMP;
        float l2 = d2 * inv[id.z] * INV_TEMP;
        float l3 = d3 * inv[id.w] * INV_TEMP;
        loss[g]  = w.x*(L - l0) + w.y*(L - l1) + w.z*(L - l2) + w.w*(L - l3);
    }
}

// ---------------- K6: final masked-mean reduction ----------------
__global__ __launch_bounds__(1024) void k_reduce(const float* __restrict__ loss,
                                                 const int*   __restrict__ kps_mask,
                                                 float* __restrict__ out) {
    __shared__ float sl[1024];
    __shared__ float sc[1024];
    int tid = threadIdx.x;
    float a = 0.f, c = 0.f;
    for (int i = tid; i < NPAIR; i += 1024) {
        a += loss[i];
        c += (kps_mask[i] != 0) ? 1.f : 0.f;
    }
    sl[tid] = a; sc[tid] = c;
    __syncthreads();
    for (int s = 512; s > 0; s >>= 1) {
        if (tid < s) { sl[tid] += sl[tid + s]; sc[tid] += sc[tid + s]; }
        __syncthreads();
    }
    if (tid == 0) out[0] = sl[0] / fmaxf(sc[0], 1.0f);
}

// ---------------- launch ----------------
extern "C" void kernel_launch(void* const* d_in, const int* in_sizes, int n_in,
                              void* d_out, int out_size, void* d_ws, size_t ws_size,
                              hipStream_t stream) {
    const float* feats_src = (const float*)d_in[0];
    const float* feats_trg = (const float*)d_in[1];
    const float* kps_src   = (const float*)d_in[2];
    const float* kps_trg   = (const float*)d_in[3];
    const int*   kps_mask  = (const int*)d_in[4];

    float* fws     = (float*)d_ws;
    float* inv_src = fws;                        // 131072 floats
    float* inv_trg = inv_src + Bsz * Mpix;       // 131072
    float* qbuf    = inv_trg + Bsz * Mpix;       // 1048576
    float* lse     = qbuf + (size_t)NPAIR * Cch; // 4096
    float* loss    = lse + NPAIR;                // 4096
    float4* srcW   = (float4*)(loss + NPAIR);    // 4096 float4
    float4* trgW   = srcW + NPAIR;
    int4*   srcIdx = (int4*)(trgW + NPAIR);
    int4*   trgIdx = srcIdx + NPAIR;

    k_invnorm<<<(Bsz * Mpix + 255) / 256, 256, 0, stream>>>(feats_src, inv_src);
    k_invnorm<<<(Bsz * Mpix + 255) / 256, 256, 0, stream>>>(feats_trg, inv_trg);
    k_prep<<<(NPAIR + 255) / 256, 256, 0, stream>>>(kps_src, kps_trg, kps_mask, inv_src,
                                                    srcW, srcIdx, trgW, trgIdx);
    k_buildq<<<NPAIR, Cch, 0, stream>>>(feats_src, srcW, srcIdx, qbuf);
    k_gemm_lse<<<Bsz * (Nkp / 16), 256, 0, stream>>>(feats_trg, inv_trg, qbuf, lse);
    k_loss<<<NPAIR / 8, 256, 0, stream>>>(feats_trg, inv_trg, qbuf, lse, trgW, trgIdx, loss);
    k_reduce<<<1, 1024, 0, stream>>>(loss, kps_mask, (float*)d_out);
}